// Decoder_56745107915335
// MI455X (gfx1250) — compile-verified
//
#include <hip/hip_runtime.h>
#include <hip/hip_bf16.h>

// ---------------------------------------------------------------------------
// Decoder with matrix recurrence, cross-attention, LN, vocab projection.
// B=32, T=64, SC=64, N=M=128, VOCAB=8192.  All fp32; matrix ops use
// V_WMMA_F32_16X16X4_F32 (the CDNA5 fp32 WMMA path), 32x32 register
// blocking per wave (4 fragments -> 4 WMMAs per K-step, 8 FLOP/byte).
// ---------------------------------------------------------------------------

#define Bsz 32
#define Tlen 64
#define SClen 64
#define Ndim 128
#define VOCAB 8192

typedef float v2f __attribute__((ext_vector_type(2)));
typedef float v8f __attribute__((ext_vector_type(8)));

// Store one 16x16 C tile from a WMMA accumulator.
// C/D layout (ISA 7.12.2): VGPR i -> row row0+i (lanes 0-15), row0+i+8 (16-31)
__device__ __forceinline__ void store_tile(float* __restrict__ Cb, const v8f& acc,
                                           int row0, int col, int ldc,
                                           float bb, int accumulate, int half)
{
#pragma unroll
    for (int i = 0; i < 8; i++) {
        const int rr = row0 + i + 8 * half;
        const size_t idx = (size_t)rr * ldc + col;
        float v = acc[i] + bb;
        if (accumulate) v += Cb[idx];
        Cb[idx] = v;
    }
}

// ---------------------------------------------------------------------------
// Generic batched GEMM:  C[b] = A[b] * op(B[b]) (+ bias) (+= C)
// Each wave computes a 32x32 tile of C (2 A-frags x 2 B-frags -> 4 WMMAs
// per K-step of 4). Block = 2 waves -> 32x64 of C per block.
// Fragment layouts per CDNA5 ISA 7.12.2 (wave32):
//   A 16x4 : lane 0-15 -> M=lane, VGPR0=K0|K2 (halves), VGPR1=K1|K3
//   B 4x16 : lane 0-15 -> N=lane, VGPR0=K0|K2,          VGPR1=K1|K3
// ---------------------------------------------------------------------------
__global__ void gemm_wmma_f32(const float* __restrict__ A,
                              const float* __restrict__ B,
                              const float* __restrict__ bias,
                              float* __restrict__ C,
                              int K, int lda, int ldb, int ldc,
                              long sA, long sB, long sC,
                              int transB, int accumulate)
{
    const int wave = threadIdx.x >> 5;
    const int lane = threadIdx.x & 31;
    const int half = lane >> 4;
    const int lid  = lane & 15;
    const long b   = blockIdx.z;

    const float* Ab = A + b * sA;
    const float* Bb = B + b * sB;
    float*       Cb = C + b * sC;

    const int row0 = blockIdx.y * 32;                // 32 rows per block
    const int col0 = blockIdx.x * 64 + wave * 32;    // 32 cols per wave
    const int bc0 = col0 + lid;
    const int bc1 = col0 + 16 + lid;

    v8f acc00 = {0.f,0.f,0.f,0.f,0.f,0.f,0.f,0.f};
    v8f acc01 = acc00, acc10 = acc00, acc11 = acc00;

    const float* arow0 = Ab + (size_t)(row0 + lid)      * lda + 2 * half;
    const float* arow1 = Ab + (size_t)(row0 + 16 + lid) * lda + 2 * half;

    if (transB) {
        const float* brow0 = Bb + (size_t)bc0 * ldb + 2 * half;
        const float* brow1 = Bb + (size_t)bc1 * ldb + 2 * half;
#pragma unroll 2
        for (int k = 0; k < K; k += 4) {
            v2f a0, a1, b0, b1;
            a0.x = arow0[k]; a0.y = arow0[k + 1];
            a1.x = arow1[k]; a1.y = arow1[k + 1];
            b0.x = brow0[k]; b0.y = brow0[k + 1];
            b1.x = brow1[k]; b1.y = brow1[k + 1];
            acc00 = __builtin_amdgcn_wmma_f32_16x16x4_f32(false, a0, false, b0, (short)0, acc00, false, false);
            acc01 = __builtin_amdgcn_wmma_f32_16x16x4_f32(false, a0, false, b1, (short)0, acc01, false, false);
            acc10 = __builtin_amdgcn_wmma_f32_16x16x4_f32(false, a1, false, b0, (short)0, acc10, false, false);
            acc11 = __builtin_amdgcn_wmma_f32_16x16x4_f32(false, a1, false, b1, (short)0, acc11, false, false);
        }
    } else {
        const float* bcol0 = Bb + (size_t)(2 * half) * ldb + bc0;
        const float* bcol1 = Bb + (size_t)(2 * half) * ldb + bc1;
#pragma unroll 2
        for (int k = 0; k < K; k += 4) {
            v2f a0, a1, b0, b1;
            a0.x = arow0[k]; a0.y = arow0[k + 1];
            a1.x = arow1[k]; a1.y = arow1[k + 1];
            b0.x = bcol0[(size_t)k * ldb]; b0.y = bcol0[(size_t)k * ldb + ldb];
            b1.x = bcol1[(size_t)k * ldb]; b1.y = bcol1[(size_t)k * ldb + ldb];
            acc00 = __builtin_amdgcn_wmma_f32_16x16x4_f32(false, a0, false, b0, (short)0, acc00, false, false);
            acc01 = __builtin_amdgcn_wmma_f32_16x16x4_f32(false, a0, false, b1, (short)0, acc01, false, false);
            acc10 = __builtin_amdgcn_wmma_f32_16x16x4_f32(false, a1, false, b0, (short)0, acc10, false, false);
            acc11 = __builtin_amdgcn_wmma_f32_16x16x4_f32(false, a1, false, b1, (short)0, acc11, false, false);
        }
    }

    const float bb0 = bias ? bias[bc0] : 0.0f;
    const float bb1 = bias ? bias[bc1] : 0.0f;
    store_tile(Cb, acc00, row0,      bc0, ldc, bb0, accumulate, half);
    store_tile(Cb, acc01, row0,      bc1, ldc, bb1, accumulate, half);
    store_tile(Cb, acc10, row0 + 16, bc0, ldc, bb0, accumulate, half);
    store_tile(Cb, acc11, row0 + 16, bc1, ldc, bb1, accumulate, half);
}

// ---------------------------------------------------------------------------
// Invert W_cell (256x256) by unpivoted Gauss-Jordan in global scratch.
// W = I + 0.02*noise -> strongly diagonal, no pivoting needed.
// One block of 256 threads, thread r owns row r of the [W | I] augment.
// ---------------------------------------------------------------------------
__global__ void invert_w_kernel(const float* __restrict__ W,
                                float* __restrict__ aug,   // 256 x 512
                                float* __restrict__ invW)  // 256 x 256
{
    const int tid = threadIdx.x;
    for (int j = 0; j < 256; j++) aug[tid * 512 + j] = W[tid * 256 + j];
    for (int j = 0; j < 256; j++) aug[tid * 512 + 256 + j] = (tid == j) ? 1.f : 0.f;
    __threadfence();
    __syncthreads();
    for (int k = 0; k < 256; k++) {
        const float pk = aug[k * 512 + k];
        if (tid != k) {
            const float f = aug[tid * 512 + k] / pk;
            for (int j = k; j < 512; j++)
                aug[tid * 512 + j] -= f * aug[k * 512 + j];
        }
        __threadfence();
        __syncthreads();
    }
    const float d = aug[tid * 512 + tid];
    for (int j = 0; j < 256; j++)
        invW[tid * 256 + j] = aug[tid * 512 + 256 + j] / d;
}

// ---------------------------------------------------------------------------
// Embedding gather: x[row, c] = emb[tokens[row], c]
// ---------------------------------------------------------------------------
__global__ void gather_kernel(const float* __restrict__ emb,
                              const int* __restrict__ tokens,
                              float* __restrict__ x)
{
    const int i = blockIdx.x * blockDim.x + threadIdx.x;
    if (i >= Bsz * Tlen * Ndim) return;
    const int row = i >> 7, c = i & 127;
    x[i] = emb[(size_t)tokens[row] * Ndim + c];
}

// A := I (per batch), alpha := 0
__global__ void init_state_kernel(float* __restrict__ A, float* __restrict__ alpha)
{
    const int i = blockIdx.x * blockDim.x + threadIdx.x;
    if (i < Bsz * Ndim * Ndim) {
        const int r = (i >> 7) & 127, c = i & 127;
        A[i] = (r == c) ? 1.f : 0.f;
    }
    if (i < Bsz * Ndim) alpha[i] = 0.f;
}

// ---------------------------------------------------------------------------
// Per-step per-batch:  solve A z = fv  (Gauss-Jordan in LDS, 128x129 fp32,
// 66 KB of CDNA5's 320KB/WG LDS), then alpha += z,
// gamma = [alpha;beta]^T * invW, carry alpha = gamma[:128], h_t = gamma[128:].
// One block of 128 threads per batch element; thread owns one matrix row.
// LDS row pad 129 -> bank = (r + j) mod 64, conflict-free.
// ---------------------------------------------------------------------------
__global__ void solve_gamma_kernel(const float* __restrict__ Acur,  // (B,128,128)
                                   const float* __restrict__ fvAll, // (B*T,128)
                                   const float* __restrict__ betaAll,
                                   float* __restrict__ alpha,       // (B,128) in/out
                                   const float* __restrict__ invW,  // 256x256
                                   float* __restrict__ h,           // (B*T,128)
                                   int t)
{
    __shared__ float aug[128][129];
    __shared__ float u[256];
    const int b = blockIdx.x;
    const int tid = threadIdx.x; // 0..127
    const float* Ab = Acur + (size_t)b * Ndim * Ndim;
    const size_t rowBT = (size_t)b * Tlen + t;

    for (int j = 0; j < 128; j++) aug[tid][j] = Ab[tid * 128 + j];
    aug[tid][128] = fvAll[rowBT * Ndim + tid];
    __syncthreads();

    for (int k = 0; k < 128; k++) {
        const float pk = aug[k][k];          // broadcast read of pivot row
        if (tid != k) {
            const float f = aug[tid][k] / pk;
            for (int j = k; j <= 128; j++)
                aug[tid][j] -= f * aug[k][j];
        }
        __syncthreads();
    }

    const float z = aug[tid][128] / aug[tid][tid];
    const float anew = alpha[b * Ndim + tid] + z;
    u[tid]       = anew;
    u[128 + tid] = betaAll[rowBT * Ndim + tid];
    __syncthreads();

    // gamma[j] = sum_i u[i] * invW[i][j]   (gamma = W^{-T} u, row form)
    float g0 = 0.f, g1 = 0.f;
    for (int i = 0; i < 256; i++) {
        const float ui = u[i];
        g0 += ui * invW[i * 256 + tid];
        g1 += ui * invW[i * 256 + tid + 128];
    }
    alpha[b * Ndim + tid] = g0;              // alpha_next = gamma[:128]
    h[rowBT * Ndim + tid] = g1;              // y_t = gamma[128:]
}

// ---------------------------------------------------------------------------
// Softmax over rows of 64 (scores), scaled by 1/sqrt(128). One wave per row.
// ---------------------------------------------------------------------------
__global__ void softmax_kernel(float* __restrict__ sc)
{
    const int wave = threadIdx.x >> 5, lane = threadIdx.x & 31;
    const int row = blockIdx.x * 4 + wave;
    float* r = sc + (size_t)row * SClen;
    const float scale = 0.08838834764831845f; // 1/sqrt(128)
    float a = r[lane] * scale, b = r[lane + 32] * scale;
    float m = fmaxf(a, b);
    for (int o = 16; o; o >>= 1) m = fmaxf(m, __shfl_xor(m, o, 32));
    const float ea = __expf(a - m), eb = __expf(b - m);
    float s = ea + eb;
    for (int o = 16; o; o >>= 1) s += __shfl_xor(s, o, 32);
    const float inv = 1.f / s;
    r[lane] = ea * inv;
    r[lane + 32] = eb * inv;
}

// ---------------------------------------------------------------------------
// LayerNorm over last dim (128), eps=1e-3, population variance.
// One wave per row, 8 rows per 256-thread block.
// ---------------------------------------------------------------------------
__global__ void layernorm_kernel(float* __restrict__ h,
                                 const float* __restrict__ g,
                                 const float* __restrict__ bta)
{
    const int wave = threadIdx.x >> 5, lane = threadIdx.x & 31;
    const int row = blockIdx.x * 8 + wave;
    float* r = h + (size_t)row * Ndim;
    float v[4], s = 0.f, s2 = 0.f;
#pragma unroll
    for (int i = 0; i < 4; i++) {
        v[i] = r[lane + 32 * i];
        s += v[i]; s2 += v[i] * v[i];
    }
    for (int o = 16; o; o >>= 1) {
        s  += __shfl_xor(s, o, 32);
        s2 += __shfl_xor(s2, o, 32);
    }
    const float mu = s * (1.f / 128.f);
    const float var = s2 * (1.f / 128.f) - mu * mu;
    const float inv = rsqrtf(var + 0.001f);
#pragma unroll
    for (int i = 0; i < 4; i++) {
        const int c = lane + 32 * i;
        r[c] = (v[i] - mu) * inv * g[c] + bta[c];
    }
}

// ---------------------------------------------------------------------------
// Host orchestration
// ---------------------------------------------------------------------------
static inline void launch_gemm(hipStream_t s,
                               const float* A, const float* B, const float* bias,
                               float* C, int M, int N, int K,
                               int lda, int ldb, int ldc,
                               long sA, long sB, long sC,
                               int batch, int transB, int acc)
{
    dim3 grid(N / 64, M / 32, batch);   // 32x64 of C per 2-wave block
    gemm_wmma_f32<<<grid, 64, 0, s>>>(A, B, bias, C, K, lda, ldb, ldc,
                                      sA, sB, sC, transB, acc);
}

extern "C" void kernel_launch(void* const* d_in, const int* in_sizes, int n_in,
                              void* d_out, int out_size, void* d_ws, size_t ws_size,
                              hipStream_t stream)
{
    (void)in_sizes; (void)n_in; (void)out_size; (void)ws_size;
    const float* context = (const float*)d_in[0];
    const int*   tokens  = (const int*)  d_in[1];
    const float* emb     = (const float*)d_in[2];
    const float* W       = (const float*)d_in[3];   // (256,256)
    const float* f_w     = (const float*)d_in[4];
    const float* f_b     = (const float*)d_in[5];
    const float* g_w     = (const float*)d_in[6];
    const float* g_b     = (const float*)d_in[7];
    const float* r_w     = (const float*)d_in[8];   // (128,16384)
    const float* r_b     = (const float*)d_in[9];
    const float* wq      = (const float*)d_in[10];
    const float* bq      = (const float*)d_in[11];
    const float* wk      = (const float*)d_in[12];
    const float* bk      = (const float*)d_in[13];
    const float* wv      = (const float*)d_in[14];
    const float* bv      = (const float*)d_in[15];
    const float* wo      = (const float*)d_in[16];
    const float* bo      = (const float*)d_in[17];
    const float* ln_g    = (const float*)d_in[18];
    const float* ln_b    = (const float*)d_in[19];
    const float* out_w   = (const float*)d_in[20];  // (128,8192)
    const float* out_b   = (const float*)d_in[21];
    float* out = (float*)d_out;

    // ---- workspace carve (~25 MB) ----
    char* p = (char*)d_ws;
    auto alloc = [&](size_t nfloats) {
        float* r = (float*)p;
        p += ((nfloats * sizeof(float) + 255) / 256) * 256;
        return r;
    };
    float* Waug  = alloc(256 * 512);
    float* invW  = alloc(256 * 256);
    float* x     = alloc(Bsz * Tlen * Ndim);
    float* fvA   = alloc(Bsz * Tlen * Ndim);
    float* betaA = alloc(Bsz * Tlen * Ndim);
    float* S     = alloc(Bsz * Ndim * Ndim);
    float* Bm    = alloc(Bsz * Ndim * Ndim);
    float* Tb    = alloc(Bsz * 256 * Ndim);
    float* A0    = alloc(Bsz * Ndim * Ndim);
    float* A1    = alloc(Bsz * Ndim * Ndim);
    float* alpha = alloc(Bsz * Ndim);
    float* h     = alloc(Bsz * Tlen * Ndim);
    float* q     = alloc(Bsz * Tlen * Ndim);
    float* kb    = alloc(Bsz * SClen * Ndim);
    float* vbuf  = alloc(Bsz * SClen * Ndim);
    float* sc    = alloc(Bsz * Tlen * SClen);
    float* ob    = alloc(Bsz * Tlen * Ndim);

    const int BT = Bsz * Tlen; // 2048

    // W^{-1} once
    invert_w_kernel<<<1, 256, 0, stream>>>(W, Waug, invW);
    // x = emb[tokens]
    gather_kernel<<<(BT * Ndim + 255) / 256, 256, 0, stream>>>(emb, tokens, x);
    // fv = x@f_w + f_b ; beta = x@g_w + g_b  (all timesteps at once)
    launch_gemm(stream, x, f_w, f_b, fvA,   BT, Ndim, Ndim, Ndim, Ndim, Ndim, 0, 0, 0, 1, 0, 0);
    launch_gemm(stream, x, g_w, g_b, betaA, BT, Ndim, Ndim, Ndim, Ndim, Ndim, 0, 0, 0, 1, 0, 0);
    // A = I, alpha = 0
    init_state_kernel<<<(Bsz * Ndim * Ndim + 255) / 256, 256, 0, stream>>>(A0, alpha);

    float* Acur = A0;
    float* Anxt = A1;
    for (int t = 0; t < Tlen; t++) {
        const float* Xt = x + (size_t)t * Ndim;     // rows stride T*N across batch
        // S = X_t @ r_w + r_b                      (32 x 16384, K=128)
        launch_gemm(stream, Xt, r_w, r_b, S,
                    Bsz, Ndim * Ndim, Ndim,
                    Tlen * Ndim, Ndim * Ndim, Ndim * Ndim,
                    0, 0, 0, 1, 0, 0);
        // Bm = S @ S^T  (batched 128x128, K=128)
        launch_gemm(stream, S, S, nullptr, Bm,
                    Ndim, Ndim, Ndim, Ndim, Ndim, Ndim,
                    Ndim * Ndim, Ndim * Ndim, Ndim * Ndim, Bsz, 1, 0);
        // T_top = A @ W11^T   (W11 = W[:128,:128], row stride 256)
        launch_gemm(stream, Acur, W, nullptr, Tb,
                    Ndim, Ndim, Ndim, Ndim, 256, Ndim,
                    Ndim * Ndim, 0, 256 * Ndim, Bsz, 1, 0);
        // T_bot = Bm @ W12^T  (W12 = W[:128,128:], base W+128)
        launch_gemm(stream, Bm, W + 128, nullptr, Tb + Ndim * Ndim,
                    Ndim, Ndim, Ndim, Ndim, 256, Ndim,
                    Ndim * Ndim, 0, 256 * Ndim, Bsz, 1, 0);
        // A_next = P @ T  where P = [W11 W12] = W[:128,:]  (K=256)
        launch_gemm(stream, W, Tb, nullptr, Anxt,
                    Ndim, Ndim, 256, 256, Ndim, Ndim,
                    0, 256 * Ndim, Ndim * Ndim, Bsz, 0, 0);
        // solve A z = fv; alpha += z; gamma = W^{-T}[alpha;beta]; emit h_t
        solve_gamma_kernel<<<Bsz, 128, 0, stream>>>(Acur, fvA, betaA, alpha, invW, h, t);
        // swap carry
        float* tmp = Acur; Acur = Anxt; Anxt = tmp;
    }

    // Attention: q = h@wq+bq ; k = ctx@wk+bk ; v = ctx@wv+bv
    launch_gemm(stream, h,       wq, bq, q,    BT, Ndim, Ndim, Ndim, Ndim, Ndim, 0, 0, 0, 1, 0, 0);
    launch_gemm(stream, context, wk, bk, kb,   BT, Ndim, Ndim, Ndim, Ndim, Ndim, 0, 0, 0, 1, 0, 0);
    launch_gemm(stream, context, wv, bv, vbuf, BT, Ndim, Ndim, Ndim, Ndim, Ndim, 0, 0, 0, 1, 0, 0);
    // scores = q @ k^T (batched 64x64, K=128)
    launch_gemm(stream, q, kb, nullptr, sc,
                Tlen, SClen, Ndim, Ndim, Ndim, SClen,
                Tlen * Ndim, SClen * Ndim, Tlen * SClen, Bsz, 1, 0);
    softmax_kernel<<<BT / 4, 128, 0, stream>>>(sc);
    // ob = softmax(scores) @ v  (batched 64x128, K=64)
    launch_gemm(stream, sc, vbuf, nullptr, ob,
                Tlen, Ndim, SClen, SClen, Ndim, Ndim,
                Tlen * SClen, SClen * Ndim, Tlen * Ndim, Bsz, 0, 0);
    // h += ob @ wo + bo
    launch_gemm(stream, ob, wo, bo, h, BT, Ndim, Ndim, Ndim, Ndim, Ndim, 0, 0, 0, 1, 0, 1);
    // LayerNorm in place
    layernorm_kernel<<<BT / 8, 256, 0, stream>>>(h, ln_g, ln_b);
    // out = h @ out_w + out_b   (2048 x 8192, K=128)
    launch_gemm(stream, h, out_w, out_b, out,
                BT, VOCAB, Ndim, Ndim, VOCAB, VOCAB, 0, 0, 0, 1, 0, 0);
}